// ForgetIntegration_54090818126193
// MI455X (gfx1250) — compile-verified
//
#include <hip/hip_runtime.h>
#include <hip/hip_bf16.h>

// Problem constants (from reference)
#define BB   32
#define LL   100
#define MCC  4
#define SS   100
#define FLL  4
#define EE   128
#define OHH  35          // 30 + 5
#define NTT  420         // FL * 3 * OH
#define KD   548         // E + NT
#define TOTK 54800       // S * (E + NT)
#define NBL  (BB * LL)       // 3200
#define NBLF (NBL * MCC)     // 12800

typedef __attribute__((ext_vector_type(2))) float v2f;
typedef __attribute__((ext_vector_type(8))) float v8f;

// ---------------------------------------------------------------------------
// Kernel 0: zero the contribution buffer + group counters
// ---------------------------------------------------------------------------
__global__ void fi_zero_kernel(float* __restrict__ contrib, int* __restrict__ counts) {
    int i = blockIdx.x * blockDim.x + threadIdx.x;
    if (i < NBLF * EE) contrib[i] = 0.0f;
    if (i < 256) counts[i] = 0;
}

// ---------------------------------------------------------------------------
// Kernel 1: bin active (bl, f) entries by slot s (last-f-wins, drop c < 0)
// ---------------------------------------------------------------------------
__global__ void fi_bin_kernel(const int* __restrict__ concepts,
                              int* __restrict__ counts, int* __restrict__ items) {
    int bl = blockIdx.x * blockDim.x + threadIdx.x;
    if (bl >= NBL) return;
    int c[MCC];
#pragma unroll
    for (int f = 0; f < MCC; ++f) c[f] = concepts[bl * MCC + f];
#pragma unroll
    for (int f = 0; f < MCC; ++f) {
        int cf = c[f];
        if (cf < 0) continue;
        bool overwritten = false;
        for (int f2 = f + 1; f2 < MCC; ++f2)
            if (c[f2] == cf) overwritten = true;
        if (overwritten) continue;
        int pos = atomicAdd(&counts[cf], 1);
        items[cf * NBL + pos] = bl * MCC + f;   // packed (bl, f)
    }
}

// ---------------------------------------------------------------------------
// Kernel 2: per active (bl,f): theta row (E) and one-hot gather contribution
//   one block per (bl,f), 128 threads (one per e). Uniform early-out.
// ---------------------------------------------------------------------------
__global__ void fi_feat_kernel(const int* __restrict__ concepts,
                               const int* __restrict__ rgaps,
                               const int* __restrict__ pcounts,
                               const int* __restrict__ acounts,
                               const float* __restrict__ emb_table,
                               const float* __restrict__ cemb_w,
                               const float* __restrict__ lstm_w,
                               float* __restrict__ theta,
                               float* __restrict__ contrib) {
    int id = blockIdx.x;          // bl*4 + f
    int bl = id >> 2;
    int f  = id & 3;
    int cf = concepts[id];
    if (cf < 0) return;
    for (int f2 = f + 1; f2 < MCC; ++f2)
        if (concepts[bl * MCC + f2] == cf) return;   // overwritten by later f

    // 12 one-hot positions inside the NT=420 feature vector
    int n[12];
    int r = rgaps[id];
#pragma unroll
    for (int fl = 0; fl < FLL; ++fl) {
        n[fl]     = fl * OHH + r;                              // oh_r (broadcast)
        n[4 + fl] = 140 + fl * OHH + pcounts[id * FLL + fl];   // oh_p
        n[8 + fl] = 280 + fl * OHH + acounts[id * FLL + fl];   // oh_a
    }

    int e = threadIdx.x;
    const float* cwr = cemb_w + (size_t)e * NTT;
    float cct = 0.0f;
#pragma unroll
    for (int j = 0; j < 12; ++j) cct += cwr[n[j]];
    theta[(size_t)id * EE + e] = emb_table[(size_t)cf * EE + e] * cct;

    const float* wr = lstm_w + (size_t)e * TOTK + (size_t)cf * KD + EE;
    float oh = 0.0f;
#pragma unroll
    for (int j = 0; j < 12; ++j) oh += wr[n[j]];
    contrib[(size_t)id * EE + e] = oh;
}

// ---------------------------------------------------------------------------
// Kernel 3: grouped GEMM per slot s using V_WMMA_F32_16X16X4_F32 (exact f32)
//   Y(group) = Theta_group (cnt x 128) x W_s^T (128 x 128)
//   one wave (32 threads) -> one 16x16 output tile, K-loop of 32 WMMAs.
// ---------------------------------------------------------------------------
__global__ __launch_bounds__(32)
void fi_gemm_kernel(const float* __restrict__ lstm_w,
                    const int* __restrict__ counts,
                    const int* __restrict__ items,
                    const float* __restrict__ theta,
                    float* __restrict__ contrib) {
    int s   = blockIdx.z;
    int cnt = counts[s];
    int mt  = blockIdx.x;
    if (mt * 16 >= cnt) return;          // uniform (scalar) branch
    int nt  = blockIdx.y;

    int t    = threadIdx.x;
    int i    = t & 15;                   // row (A) / col (B) within tile
    int half = t >> 4;                   // lane half selects K pair
    int kh   = half * 2;

    int  m     = mt * 16 + i;
    bool valid = (m < cnt);
    int  item  = valid ? items[s * NBL + m] : 0;
    float vmask = valid ? 1.0f : 0.0f;   // branchless zero for tail rows

    const float* arow = theta + (size_t)item * EE;                 // A: 16 x K
    int e = nt * 16 + i;
    const float* bcol = lstm_w + (size_t)e * TOTK + (size_t)s * KD; // B: K x 16

    v8f acc = {};
#pragma unroll
    for (int k0 = 0; k0 < EE; k0 += 4) {
        v2f a, b;
        // A layout (16x4 f32): lanes 0-15 K={0,1}, lanes 16-31 K={2,3}
        a.x = arow[k0 + kh]     * vmask;
        a.y = arow[k0 + kh + 1] * vmask;
        // B layout (4x16 f32): VGPRv lanes0-15 row v, lanes16-31 row v+2
        b.x = bcol[k0 + kh];
        b.y = bcol[k0 + kh + 1];
        acc = __builtin_amdgcn_wmma_f32_16x16x4_f32(
            /*neg_a=*/false, a, /*neg_b=*/false, b,
            /*c_mod=*/(short)0, acc, /*reuse_a=*/false, /*reuse_b=*/false);
    }

    // D layout: VGPR v holds row (v + 8*half), col i. Each (item,e) has a
    // single writer across the whole grid -> plain read-modify-write.
#pragma unroll
    for (int v = 0; v < 8; ++v) {
        int mr = mt * 16 + v + half * 8;
        if (mr < cnt) {
            int it2 = items[s * NBL + mr];
            size_t off = (size_t)it2 * EE + e;
            contrib[off] += acc[v];
        }
    }
}

// ---------------------------------------------------------------------------
// Kernel 4: out[bl, e] = lstm_b[e] + sum_f contrib[(bl,f), e]  (fixed order)
// ---------------------------------------------------------------------------
__global__ void fi_reduce_kernel(const float* __restrict__ contrib,
                                 const float* __restrict__ lstm_b,
                                 float* __restrict__ out) {
    int bl = blockIdx.x;
    int e  = threadIdx.x;
    float r = lstm_b[e];
#pragma unroll
    for (int f = 0; f < MCC; ++f)
        r += contrib[(size_t)(bl * MCC + f) * EE + e];
    out[(size_t)bl * EE + e] = r;
}

// ---------------------------------------------------------------------------
// Launcher
// ---------------------------------------------------------------------------
extern "C" void kernel_launch(void* const* d_in, const int* in_sizes, int n_in,
                              void* d_out, int out_size, void* d_ws, size_t ws_size,
                              hipStream_t stream) {
    (void)in_sizes; (void)n_in; (void)out_size; (void)ws_size;

    const int*   concepts  = (const int*)d_in[0];
    const int*   rgaps     = (const int*)d_in[1];
    const int*   pcounts   = (const int*)d_in[2];
    const int*   acounts   = (const int*)d_in[3];
    const float* emb_table = (const float*)d_in[4];
    const float* cemb_w    = (const float*)d_in[5];
    const float* lstm_w    = (const float*)d_in[6];
    const float* lstm_b    = (const float*)d_in[7];
    float*       out       = (float*)d_out;

    // Workspace layout (~14.4 MB):
    int*   counts  = (int*)d_ws;                       // 256 ints (100 used)
    int*   items   = counts + 256;                     // S * NBL ints
    float* theta   = (float*)(items + SS * NBL);       // NBLF * E floats
    float* contrib = theta + (size_t)NBLF * EE;        // NBLF * E floats

    fi_zero_kernel<<<(NBLF * EE) / 256, 256, 0, stream>>>(contrib, counts);
    fi_bin_kernel<<<(NBL + 255) / 256, 256, 0, stream>>>(concepts, counts, items);
    fi_feat_kernel<<<NBLF, EE, 0, stream>>>(concepts, rgaps, pcounts, acounts,
                                            emb_table, cemb_w, lstm_w,
                                            theta, contrib);
    dim3 gg(NBL / 16, EE / 16, SS);   // (200, 8, 100); most tiles exit early
    fi_gemm_kernel<<<gg, 32, 0, stream>>>(lstm_w, counts, items, theta, contrib);
    fi_reduce_kernel<<<NBL, EE, 0, stream>>>(contrib, lstm_b, out);
}